// GCNClustering_12240656794220
// MI455X (gfx1250) — compile-verified
//
#include <hip/hip_runtime.h>

#define IN_CH  128
#define HID    64
#define OUT_CH 16

typedef float v2f __attribute__((ext_vector_type(2)));
typedef float v8f __attribute__((ext_vector_type(8)));

// ---------------------------------------------------------------------------
// init: deg[i]=1 (self-loop), agg1 (h) = b1 broadcast, out = b2 broadcast
// ---------------------------------------------------------------------------
__global__ void gcn_init_kernel(float* __restrict__ deg,
                                float* __restrict__ agg1, const float* __restrict__ b1,
                                float* __restrict__ out,  const float* __restrict__ b2,
                                int nNodes) {
    long long idx = (long long)blockIdx.x * blockDim.x + threadIdx.x;
    if (idx >= (long long)nNodes * HID) return;
    int i = (int)(idx >> 6);          // HID == 64
    int c = (int)(idx & (HID - 1));
    agg1[idx] = b1[c];
    if (c == 0)     deg[i] = 1.0f;                    // self-loop contribution
    if (c < OUT_CH) out[(long long)i * OUT_CH + c] = b2[c];
}

// ---------------------------------------------------------------------------
// degree accumulation over real edges (self-loop already counted in init)
// ---------------------------------------------------------------------------
__global__ void gcn_degree_kernel(const int* __restrict__ dst,
                                  float* __restrict__ deg, int nEdges) {
    int e = blockIdx.x * blockDim.x + threadIdx.x;
    if (e < nEdges) atomicAdd(&deg[dst[e]], 1.0f);
}

__global__ void gcn_rsqrt_kernel(float* __restrict__ deg, int n) {
    int i = blockIdx.x * blockDim.x + threadIdx.x;
    if (i < n) deg[i] = rsqrtf(deg[i]);   // deg >= 1 always (self loops)
}

// ---------------------------------------------------------------------------
// C[M,N] = A[M,K] x W[K,N], fp32, via V_WMMA_F32_16X16X4_F32.
// One wave computes one 16x16 output tile; K consumed 4 at a time.
// A 16x4 layout : lane m=lane&15 holds row m; lanes<16 -> K k+0,k+1 in the
//                 two VGPRs, lanes>=16 -> K k+2,k+3.
// B 4x16 layout : lane n=lane&15 holds col n; same K split across lane halves.
// C/D layout    : VGPR r, lane l -> row r + 8*(l>=16), col l&15.
// ---------------------------------------------------------------------------
__global__ void gcn_gemm_wmma_f32(const float* __restrict__ A,
                                  const float* __restrict__ W,
                                  float* __restrict__ C,
                                  int M, int K, int N) {
    const int wavesPerBlock = blockDim.x >> 5;
    const int wave = (int)(threadIdx.x >> 5);
    const int lane = (int)(threadIdx.x & 31);
    const int tilesN = N >> 4;
    const int gw = blockIdx.x * wavesPerBlock + wave;
    const int tileM = gw / tilesN;
    const int tileN = gw - tileM * tilesN;
    if (tileM * 16 >= M) return;          // wave-uniform: EXEC stays all-ones

    const int row0 = tileM << 4;
    const int col0 = tileN << 4;
    const int mr = lane & 15;             // A row / B col owned by this lane
    const int kh = (lane >> 4) << 1;      // 0 for lanes 0-15, 2 for lanes 16-31

    v8f acc = {};
    for (int k = 0; k < K; k += 4) {
        v2f a, b;
        const float* ap = A + (long long)(row0 + mr) * K + (k + kh);
        a.x = ap[0];
        a.y = ap[1];
        const float* bp = W + (long long)(k + kh) * N + (col0 + mr);
        b.x = bp[0];
        b.y = bp[N];
        acc = __builtin_amdgcn_wmma_f32_16x16x4_f32(
            /*neg_a=*/false, a, /*neg_b=*/false, b,
            /*c_mod=*/(short)0, acc, /*reuse_a=*/false, /*reuse_b=*/false);
    }

    const int crow = row0 + ((lane >> 4) << 3);
    const int ccol = col0 + mr;
#pragma unroll
    for (int r = 0; r < 8; ++r)
        C[(long long)(crow + r) * N + ccol] = acc[r];
}

// ---------------------------------------------------------------------------
// Edge-parallel normalized scatter-add: one thread per (edge, channel).
// Edges [0,nEdges) come from the index arrays; edges [nEdges, nEdges+nNodes)
// are the self-loops (s == d == e - nEdges).
// ---------------------------------------------------------------------------
template <int C>
__global__ void gcn_scatter_kernel(const float* __restrict__ xl,
                                   const int* __restrict__ src,
                                   const int* __restrict__ dst,
                                   const float* __restrict__ dinv,
                                   float* __restrict__ agg,
                                   int nEdges, int nNodes) {
    long long idx = (long long)blockIdx.x * blockDim.x + threadIdx.x;
    long long total = (long long)(nEdges + nNodes) * C;
    if (idx >= total) return;
    int c = (int)(idx % C);
    long long e = idx / C;
    int s, d;
    if (e < nEdges) { s = src[e]; d = dst[e]; }
    else            { s = d = (int)(e - nEdges); }
    float norm = dinv[s] * dinv[d];
    float v = xl[(long long)s * C + c] * norm;
    atomicAdd(&agg[(long long)d * C + c], v);
}

__global__ void gcn_relu_kernel(float* __restrict__ h, long long n) {
    long long i = (long long)blockIdx.x * blockDim.x + threadIdx.x;
    if (i < n) h[i] = fmaxf(h[i], 0.0f);
}

// ---------------------------------------------------------------------------
extern "C" void kernel_launch(void* const* d_in, const int* in_sizes, int n_in,
                              void* d_out, int out_size, void* d_ws, size_t ws_size,
                              hipStream_t stream) {
    const float* x  = (const float*)d_in[0];
    const int*   ei = (const int*)  d_in[1];   // [2, E] flattened: row0=src, row1=dst
    const float* W1 = (const float*)d_in[2];
    const float* b1 = (const float*)d_in[3];
    const float* W2 = (const float*)d_in[4];
    const float* b2 = (const float*)d_in[5];
    float* out = (float*)d_out;

    const int nNodes = in_sizes[0] / IN_CH;
    const int nEdges = in_sizes[1] / 2;
    const int* src = ei;
    const int* dst = ei + nEdges;

    // workspace layout (floats): deg | xl1[N,64] | h[N,64] | xl2[N,16]
    float* ws  = (float*)d_ws;
    float* deg = ws;                                   // also holds dinv after rsqrt
    float* xl1 = deg + nNodes;
    float* h   = xl1 + (long long)nNodes * HID;
    float* xl2 = h   + (long long)nNodes * HID;

    const int B = 256;
    const long long hElems = (long long)nNodes * HID;

    // 1) init accumulators with biases, deg with self-loop count
    gcn_init_kernel<<<(int)((hElems + B - 1) / B), B, 0, stream>>>(deg, h, b1, out, b2, nNodes);

    // 2) degree accumulation + 3) dinv = rsqrt(deg)
    gcn_degree_kernel<<<(nEdges + B - 1) / B, B, 0, stream>>>(dst, deg, nEdges);
    gcn_rsqrt_kernel<<<(nNodes + B - 1) / B, B, 0, stream>>>(deg, nNodes);

    // 4) xl1 = x @ W1   (fp32 WMMA)
    {
        int tiles = (nNodes / 16) * (HID / 16);
        int wpb = B / 32;
        gcn_gemm_wmma_f32<<<(tiles + wpb - 1) / wpb, B, 0, stream>>>(x, W1, xl1, nNodes, IN_CH, HID);
    }

    // 5) layer-1 normalized scatter-add into h (pre-initialized with b1)
    {
        long long tot = (long long)(nEdges + nNodes) * HID;
        gcn_scatter_kernel<HID><<<(int)((tot + B - 1) / B), B, 0, stream>>>(
            xl1, src, dst, deg, h, nEdges, nNodes);
    }

    // 6) ReLU in place
    gcn_relu_kernel<<<(int)((hElems + B - 1) / B), B, 0, stream>>>(h, hElems);

    // 7) xl2 = h @ W2   (fp32 WMMA)
    {
        int tiles = (nNodes / 16) * (OUT_CH / 16);
        int wpb = B / 32;
        gcn_gemm_wmma_f32<<<(tiles + wpb - 1) / wpb, B, 0, stream>>>(h, W2, xl2, nNodes, HID, OUT_CH);
    }

    // 8) layer-2 normalized scatter-add into out (pre-initialized with b2)
    {
        long long tot = (long long)(nEdges + nNodes) * OUT_CH;
        gcn_scatter_kernel<OUT_CH><<<(int)((tot + B - 1) / B), B, 0, stream>>>(
            xl2, src, dst, deg, out, nEdges, nNodes);
    }
}